// BoxAttention_56573309224536
// MI455X (gfx1250) — compile-verified
//
#include <hip/hip_runtime.h>
#include <math.h>

// ---- problem constants (from reference) ----
#define BATCH      4
#define D_MODEL    256
#define NUM_HEADS  8
#define HEAD_DIM   32
#define LTOT       13294              // 100*100 + 50*50 + 25*25 + 13*13
#define MREAL      (BATCH * LTOT)     // 53176
#define MPAD       53184              // ceil(MREAL/16)*16

typedef __attribute__((ext_vector_type(2))) float v2f;
typedef __attribute__((ext_vector_type(8))) float v8f;

// =====================================================================
// 1) Flatten (B,256,H,W) levels -> (Mpad,256) row-major; query = value+pos
//    Padded rows are zeroed so WMMA tiles need no edge handling.
// =====================================================================
__global__ void flatten_kernel(const float* __restrict__ v0, const float* __restrict__ v1,
                               const float* __restrict__ v2, const float* __restrict__ v3,
                               const float* __restrict__ p0, const float* __restrict__ p1,
                               const float* __restrict__ p2, const float* __restrict__ p3,
                               float* __restrict__ vflat, float* __restrict__ qflat)
{
    long long t = (long long)blockIdx.x * blockDim.x + threadIdx.x;
    const long long total = (long long)MPAD * D_MODEL;
    if (t >= total) return;
    int m = (int)(t >> 8);
    int c = (int)(t & 255);
    if (m >= MREAL) { vflat[t] = 0.f; qflat[t] = 0.f; return; }
    int b = m / LTOT;
    int p = m - b * LTOT;
    const float* vsrc; const float* psrc; int HW, loc;
    if      (p < 10000) { vsrc = v0; psrc = p0; HW = 10000; loc = p;         }
    else if (p < 12500) { vsrc = v1; psrc = p1; HW = 2500;  loc = p - 10000; }
    else if (p < 13125) { vsrc = v2; psrc = p2; HW = 625;   loc = p - 12500; }
    else                { vsrc = v3; psrc = p3; HW = 169;   loc = p - 13125; }
    long long si = (long long)(b * D_MODEL + c) * HW + loc;   // (B,256,H,W) layout
    float v = vsrc[si];
    vflat[t] = v;
    qflat[t] = v + psrc[si];
}

// =====================================================================
// 2) f32 WMMA GEMM:  C[Mpad,N] = A[Mpad,256] @ W[N,256]^T + bias[N]
//    Block: 16 rows x N cols; N/16 waves.
//    A strip staged in LDS via GLOBAL_LOAD_ASYNC_TO_LDS_B128 (ASYNCcnt).
//    Each wave: 64 x V_WMMA_F32_16X16X4_F32 over K=256.
//    N is compile-time so store offsets fold to immediates; GUARD only
//    instantiated for the final GEMM (d_out is exactly MREAL rows).
// =====================================================================
template <int N, bool GUARD>
__global__ __launch_bounds__(512) void gemm_wmma_f32(const float* __restrict__ A,
                                                     const float* __restrict__ Wt,
                                                     const float* __restrict__ bias,
                                                     float* __restrict__ C,
                                                     int Mstore)
{
    __shared__ float As[16 * 256];               // only LDS object -> offset 0
    constexpr int NWAVES   = N / 16;
    constexpr int NTHREADS = NWAVES * 32;
    constexpr int NVEC     = (16 * 256) / 4;     // 1024 float4 chunks
    const int tid = threadIdx.x;
    const int m0  = blockIdx.x * 16;

    // --- async 128-bit copy of the 16x256 A strip directly into LDS ---
    {
        const float* Ablk = A + (size_t)m0 * 256;
#pragma unroll
        for (int it = 0; it < NVEC / NTHREADS; ++it) {
            unsigned byteoff = (unsigned)((tid + it * NTHREADS) * 16);
            // %0: LDS byte offset; %1: byte offset added to Ablk (%2) for the global source
            asm volatile("global_load_async_to_lds_b128 %0, %1, %2"
                         :: "v"(byteoff), "v"(byteoff), "s"(Ablk)
                         : "memory");
        }
        asm volatile("s_wait_asynccnt 0" ::: "memory");
    }
    __syncthreads();

    const int wave = tid >> 5;
    const int lane = tid & 31;
    const int col  = wave * 16 + (lane & 15);   // output column owned by this lane
    const int rowi = lane & 15;                 // A row inside the tile
    const int koff = (lane >> 4) * 2;           // K sub-offset per ISA A/B layout
    const float* __restrict__ wrow = Wt + (size_t)col * 256;
    const float bn = bias[col];

    v8f acc = {};
#pragma unroll 8
    for (int k0 = 0; k0 < 256; k0 += 4) {
        v2f a = *(const v2f*)(&As[rowi * 256 + k0 + koff]);     // 8B-aligned ds load
        v2f b = *(const v2f*)(&wrow[k0 + koff]);                // 8B-aligned global load
        acc = __builtin_amdgcn_wmma_f32_16x16x4_f32(false, a, false, b,
                                                    (short)0, acc, false, false);
    }

    const int rbase = (lane >> 4) * 8;          // lanes 16-31 hold rows M=8..15
    float* __restrict__ Crow = C + (size_t)(m0 + rbase) * N + col;
    if (!GUARD || (m0 + 16 <= Mstore)) {
#pragma unroll
        for (int r = 0; r < 8; ++r)
            Crow[(size_t)r * N] = acc[r] + bn;  // compile-time offsets
    } else {
#pragma unroll
        for (int r = 0; r < 8; ++r)
            if (m0 + rbase + r < Mstore) Crow[(size_t)r * N] = acc[r] + bn;
    }
}

// =====================================================================
// 3) softmax over the 16 attention logits per (query, head), in place
// =====================================================================
__global__ void softmax16_kernel(float* __restrict__ logits, int total)
{
    int t = blockIdx.x * blockDim.x + threadIdx.x;
    if (t >= total) return;
    float* p = logits + (long long)t * 16;
    float mx = p[0];
#pragma unroll
    for (int i = 1; i < 16; ++i) mx = fmaxf(mx, p[i]);
    float s = 0.f, e[16];
#pragma unroll
    for (int i = 0; i < 16; ++i) { e[i] = __expf(p[i] - mx); s += e[i]; }
    float inv = 1.f / s;
#pragma unroll
    for (int i = 0; i < 16; ++i) p[i] = e[i] * inv;
}

// =====================================================================
// 4) deformable bilinear sampling + attention-weighted accumulation
//    one wave per (query m, head); lane = channel d (coalesced gathers)
// =====================================================================
__global__ __launch_bounds__(256) void sample_kernel(const float* __restrict__ vp,
                                                     const float* __restrict__ attn,
                                                     const float* __restrict__ offs,
                                                     float* __restrict__ outh)
{
    const int lane = threadIdx.x & 31;
    const int m    = blockIdx.x;                 // one block = 8 waves = 8 heads of one m
    const int head = threadIdx.x >> 5;
    float* outp = outh + (long long)m * D_MODEL + head * HEAD_DIM + lane;
    if (m >= MREAL) { *outp = 0.f; return; }     // zero padded rows for final GEMM

    int b = m / LTOT;
    int p = m - b * LTOT;

    // reference window for this query's own level/position
    int Hq, Wq, locq;
    if      (p < 10000) { Hq = 100; Wq = 100; locq = p;         }
    else if (p < 12500) { Hq = 50;  Wq = 50;  locq = p - 10000; }
    else if (p < 13125) { Hq = 25;  Wq = 25;  locq = p - 12500; }
    else                { Hq = 13;  Wq = 13;  locq = p - 13125; }
    int hq = locq / Wq, wq = locq - hq * Wq;
    float rcx = ((float)wq + 0.5f) / ((float)Wq + 1e-6f);
    float rcy = ((float)hq + 0.5f) / ((float)Hq + 1e-6f);
    float rw  = 4.0f / (float)Wq;
    float rh  = 4.0f / (float)Hq;

    const float* offm = offs + (long long)m * 128 + head * 16;  // (8,4,4) layout
    const float* attm = attn + (long long)m * 128 + head * 16;

    const int Hs[4]   = {100, 50, 25, 13};
    const int lofs[4] = {0, 10000, 12500, 13125};

    float acc = 0.f;
#pragma unroll
    for (int lev = 0; lev < 4; ++lev) {
        const int H = Hs[lev], W = Hs[lev];
        const long long base = ((long long)(b * LTOT + lofs[lev])) * D_MODEL
                               + head * HEAD_DIM + lane;
        float o0 = offm[lev * 4 + 0], o1 = offm[lev * 4 + 1];
        float o2 = offm[lev * 4 + 2], o3 = offm[lev * 4 + 3];
        float cx = rcx + o0 * 0.125f * rw;
        float cy = rcy + o1 * 0.125f * rh;
        float bw = fmaxf(rw + o2 * 0.125f * rw, 0.f);   // relu(size)
        float bh = fmaxf(rh + o3 * 0.125f * rh, 0.f);
#pragma unroll
        for (int pt = 0; pt < 4; ++pt) {
            float kx = (pt & 1) ? 0.25f : -0.25f;       // kernel indices (+-0.25)
            float ky = (pt & 2) ? 0.25f : -0.25f;
            float gx = cx + kx * bw;
            float gy = cy + ky * bh;
            float x = gx * (float)W - 0.5f;
            float y = gy * (float)H - 0.5f;
            float xf = floorf(x), yf = floorf(y);
            float lx = x - xf, ly = y - yf;
            int ix = (int)xf, iy = (int)yf;
            float aw = attm[lev * 4 + pt];
            float w00 = (1.f - lx) * (1.f - ly), w10 = lx * (1.f - ly);
            float w01 = (1.f - lx) * ly,         w11 = lx * ly;
            // corner validity is wave-uniform (same m,head across lanes)
            if (ix >= 0     && ix < W     && iy >= 0     && iy < H)
                acc += aw * w00 * vp[base + (long long)(iy * W + ix) * D_MODEL];
            if (ix + 1 >= 0 && ix + 1 < W && iy >= 0     && iy < H)
                acc += aw * w10 * vp[base + (long long)(iy * W + ix + 1) * D_MODEL];
            if (ix >= 0     && ix < W     && iy + 1 >= 0 && iy + 1 < H)
                acc += aw * w01 * vp[base + (long long)((iy + 1) * W + ix) * D_MODEL];
            if (ix + 1 >= 0 && ix + 1 < W && iy + 1 >= 0 && iy + 1 < H)
                acc += aw * w11 * vp[base + (long long)((iy + 1) * W + ix + 1) * D_MODEL];
        }
    }
    *outp = acc;
}

// =====================================================================
// host launcher
// =====================================================================
extern "C" void kernel_launch(void* const* d_in, const int* in_sizes, int n_in,
                              void* d_out, int out_size, void* d_ws, size_t ws_size,
                              hipStream_t stream)
{
    const float* v0    = (const float*)d_in[0];
    const float* v1    = (const float*)d_in[1];
    const float* v2    = (const float*)d_in[2];
    const float* v3    = (const float*)d_in[3];
    const float* p0    = (const float*)d_in[4];
    const float* p1    = (const float*)d_in[5];
    const float* p2    = (const float*)d_in[6];
    const float* p3    = (const float*)d_in[7];
    const float* Wv    = (const float*)d_in[8];
    const float* bv    = (const float*)d_in[9];
    const float* Wo    = (const float*)d_in[10];
    const float* bo    = (const float*)d_in[11];
    const float* Wbox  = (const float*)d_in[12];
    const float* bbox  = (const float*)d_in[13];
    const float* Wattn = (const float*)d_in[14];
    const float* battn = (const float*)d_in[15];

    float* ws    = (float*)d_ws;
    float* qflat = ws;  ws += (size_t)MPAD * 256;
    float* vflat = ws;  ws += (size_t)MPAD * 256;
    float* vp    = ws;  ws += (size_t)MPAD * 256;
    float* attn  = ws;  ws += (size_t)MPAD * 128;
    float* offs  = ws;  ws += (size_t)MPAD * 128;
    float* outh  = ws;  ws += (size_t)MPAD * 256;

    // 1) flatten + query = value + pos (padded rows zeroed)
    {
        long long total = (long long)MPAD * 256;
        int threads = 256;
        unsigned blocks = (unsigned)((total + threads - 1) / threads);
        flatten_kernel<<<blocks, threads, 0, stream>>>(v0, v1, v2, v3, p0, p1, p2, p3,
                                                       vflat, qflat);
    }

    const int mblocks = MPAD / 16;
    // 2) three projection GEMMs (WMMA f32, unguarded: buffers are MPAD rows)
    gemm_wmma_f32<256, false><<<mblocks, 512, 0, stream>>>(vflat, Wv,    bv,    vp,   MPAD);
    gemm_wmma_f32<128, false><<<mblocks, 256, 0, stream>>>(qflat, Wattn, battn, attn, MPAD);
    gemm_wmma_f32<128, false><<<mblocks, 256, 0, stream>>>(qflat, Wbox,  bbox,  offs, MPAD);

    // 3) softmax over 16 logits per (m, head)
    {
        int total = MREAL * NUM_HEADS;
        int threads = 256;
        softmax16_kernel<<<(total + threads - 1) / threads, threads, 0, stream>>>(attn, total);
    }

    // 4) bilinear sampling + attention accumulation (also zeroes padded rows)
    sample_kernel<<<MPAD, 256, 0, stream>>>(vp, attn, offs, outh);

    // 5) output projection into d_out (store-guarded at MREAL, last block only)
    gemm_wmma_f32<256, true><<<mblocks, 512, 0, stream>>>(outh, Wo, bo, (float*)d_out, MREAL);
}